// AdditiveAttention_58282706207316
// MI455X (gfx1250) — compile-verified
//
#include <hip/hip_runtime.h>
#include <math.h>

// Problem constants (from reference)
constexpr int B = 32;
constexpr int T = 8192;
constexpr int E = 512;
constexpr int D = 256;

// Tiling
constexpr int CHUNKS = 32;          // T-chunks per batch -> 32*32 = 1024 workgroups
constexpr int CT     = T / CHUNKS;  // 256 rows per chunk
constexpr int TM     = 32;          // tile rows held in LDS
constexpr int NT     = CT / TM;     // 8 tile steps per chunk
constexpr int LDSW   = E + 4;       // padded row stride (516): conflict-free b64 patterns

typedef float v2f __attribute__((ext_vector_type(2)));
typedef float v4f __attribute__((ext_vector_type(4)));
typedef float v8f __attribute__((ext_vector_type(8)));
typedef int   v4i __attribute__((vector_size(4 * sizeof(int))));  // matches builtin param

#define GAS __attribute__((address_space(1)))
#define LAS __attribute__((address_space(3)))

// Async global->LDS staging (CDNA5): tracked by ASYNCcnt, no VGPR round-trip.
#if __has_builtin(__builtin_amdgcn_global_load_async_to_lds_b128)
#define USE_ASYNC_LDS 1
#else
#define USE_ASYNC_LDS 0
#endif

__device__ __forceinline__ void wait_async0() {
#if USE_ASYNC_LDS
#if __has_builtin(__builtin_amdgcn_s_wait_asynccnt)
    __builtin_amdgcn_s_wait_asynccnt(0);
#else
    asm volatile("s_wait_asynccnt 0x0" ::: "memory");
#endif
#endif
}

// Branch-free tanh: prefer hardware V_TANH (gfx1250 trans op), else exp+rcp.
__device__ __forceinline__ float fast_tanh(float x) {
#if __has_builtin(__builtin_amdgcn_tanhf)
    return __builtin_amdgcn_tanhf(x);
#elif __has_builtin(__builtin_amdgcn_tanh_f32)
    return __builtin_amdgcn_tanh_f32(x);
#else
    // tanh(x) = 1 - 2/(exp(2x)+1); exact at +/-inf, no branches.
    float e = __expf(2.0f * x);
    return 1.0f - 2.0f * __builtin_amdgcn_rcpf(e + 1.0f);
#endif
}

// ---------------------------------------------------------------------------
// Kernel 0: cst[b,d] = sum_k query[b,k]*W1_w[d,k] + W1_b[d] + W2_b[d]
// ---------------------------------------------------------------------------
__global__ __launch_bounds__(D) void prep_kernel(const float* __restrict__ query,
                                                 const float* __restrict__ W1_w,
                                                 const float* __restrict__ W1_b,
                                                 const float* __restrict__ W2_b,
                                                 float* __restrict__ cst) {
    __shared__ float qs[D];
    const int b = blockIdx.x;
    const int d = threadIdx.x;
    qs[d] = query[b * D + d];
    __syncthreads();
    float acc = W1_b[d] + W2_b[d];
    const float* wrow = W1_w + (size_t)d * D;
#pragma unroll 8
    for (int k = 0; k < D; ++k) acc = fmaf(qs[k], wrow[k], acc);
    cst[b * D + d] = acc;
}

// ---------------------------------------------------------------------------
// Kernel 1: fused  v_proj GEMM (WMMA f32) -> tanh -> dot v -> online softmax
//           -> weighted accumulation of values. One pass over `values`.
// Per block: (b, chunk of 256 rows). 256 threads = 8 wave32.
// ---------------------------------------------------------------------------
__global__ __launch_bounds__(256) void main_kernel(const float* __restrict__ values,
                                                   const float* __restrict__ W2_w,
                                                   const float* __restrict__ vvec,
                                                   const float* __restrict__ cst,
                                                   float* __restrict__ part_m,
                                                   float* __restrict__ part_s,
                                                   float* __restrict__ part_acc) {
    __shared__ __align__(16) float sv[2][TM][LDSW];   // 132,096 B double-buffered tile
    __shared__ float s_scores[TM];
    __shared__ float s_w[TM];
    __shared__ float s_cst[D];
    __shared__ float s_v[D];

    const int tid   = threadIdx.x;
    const int b     = blockIdx.y;
    const int chunk = blockIdx.x;
    const int tbase = chunk * CT;

    // per-batch constants into LDS (tid covers D==256)
    s_cst[tid] = cst[b * D + tid];
    s_v[tid]   = vvec[tid];

    // tile stager: 32 rows x 512 f32 per tile, 16 b128 per thread.
    auto stage = [&](int bufi, int tile) {
        const float* src = values + ((size_t)b * T + tbase + tile * TM) * E;
#pragma unroll
        for (int i = 0; i < (TM * E / 4) / 256; ++i) {   // 16 float4 per thread
            const int flat = tid + i * 256;              // float4 index
            const int r = flat >> 7;                     // E/4 = 128 float4 per row
            const int c = (flat & 127) << 2;
            const float* gp = src + (size_t)r * E + c;
            float*       lp = &sv[bufi][r][c];
#if USE_ASYNC_LDS
            __builtin_amdgcn_global_load_async_to_lds_b128(
                (GAS v4i*)(v4i*)const_cast<float*>(gp),
                (LAS v4i*)(v4i*)lp, 0, 0);
#else
            v4f val = __builtin_nontemporal_load((const v4f*)gp);
            *(v4f*)lp = val;
#endif
        }
    };

    // wave-level WMMA geometry: 8 waves cover TM(32) x D(256) output
    const int lane = tid & 31;
    const int w    = tid >> 5;
    const int rb   = w & 1;        // row block (16 rows)
    const int dq   = w >> 1;       // 64-wide d slab (4 d-tiles of 16)
    const int l16  = lane & 15;
    const int half = lane >> 4;
    const int m0   = rb * 16;
    const int d0   = dq * 64;

    float m_run = -INFINITY, s_run = 0.f, accx = 0.f, accy = 0.f;

    stage(0, 0);
    int buf = 0;
    for (int tile = 0; tile < NT; ++tile) {
        if (tid < TM) s_scores[tid] = 0.f;
        wait_async0();                                     // my copies into `buf` done
        __syncthreads();                                   // everyone's copies visible
        if (tile + 1 < NT) stage(buf ^ 1, tile + 1);       // async copy overlaps compute

        // ---- GEMM: vproj[m0..m0+15][d0..d0+63] += tile @ W2_w^T over E ----
        v8f acc[4] = {};
        // A frag: lane l16 = row, half selects K pair {0,1}/{2,3}
        const float* arow  = &sv[buf][m0 + l16][2 * half];
        // B frag: lane l16 = output col d, contiguous K along W2_w row
        const float* brow0 = W2_w + (size_t)(d0 + l16) * E + 2 * half;
#pragma unroll 4
        for (int e0 = 0; e0 < E; e0 += 4) {
            v2f a = *(const v2f*)(arow + e0);
#pragma unroll
            for (int dt = 0; dt < 4; ++dt) {
                v2f bb = *(const v2f*)(brow0 + (size_t)dt * 16 * E + e0);
                acc[dt] = __builtin_amdgcn_wmma_f32_16x16x4_f32(
                    false, a, false, bb, (short)0, acc[dt], false, false);
            }
        }

        // ---- epilogue: score[m] += sum_d tanh(vproj + cst[d]) * v[d] ----
        float p[8];
#pragma unroll
        for (int r = 0; r < 8; ++r) p[r] = 0.f;
#pragma unroll
        for (int dt = 0; dt < 4; ++dt) {
            const int d  = d0 + dt * 16 + l16;
            const float cc = s_cst[d];
            const float vv = s_v[d];
#pragma unroll
            for (int r = 0; r < 8; ++r)
                p[r] += fast_tanh(acc[dt][r] + cc) * vv;
        }
#pragma unroll
        for (int r = 0; r < 8; ++r) {
            float x = p[r];
            x += __shfl_xor(x, 1);
            x += __shfl_xor(x, 2);
            x += __shfl_xor(x, 4);
            x += __shfl_xor(x, 8);
            if (l16 == 0)   // lanes 0 and 16: rows m0+r and m0+r+8
                atomicAdd(&s_scores[m0 + r + 8 * half], x);
        }
        __syncthreads();                                   // scores complete

        // ---- online softmax update (replicated scalar state) ----
        float mt = -INFINITY;
#pragma unroll
        for (int i = 0; i < TM / 4; ++i) {
            v4f sc = *(const v4f*)&s_scores[i * 4];
            mt = fmaxf(mt, fmaxf(fmaxf(sc.x, sc.y), fmaxf(sc.z, sc.w)));
        }
        const float m_new = fmaxf(m_run, mt);
        const float corr  = __expf(m_run - m_new);
        s_run *= corr; accx *= corr; accy *= corr;
        if (tid < TM) s_w[tid] = __expf(s_scores[tid] - m_new);
        __syncthreads();
        float ssum = 0.f;
#pragma unroll
        for (int i = 0; i < TM / 4; ++i) {
            v4f w4 = *(const v4f*)&s_w[i * 4];
            ssum += (w4.x + w4.y) + (w4.z + w4.w);
        }
        s_run += ssum;
        m_run = m_new;

        // ---- weighted accumulation: thread owns e = 2*tid, 2*tid+1 ----
        const float* col = &sv[buf][0][2 * tid];
#pragma unroll 4
        for (int t = 0; t < TM; ++t) {
            const float wt = s_w[t];
            v2f vals = *(const v2f*)(col + (size_t)t * LDSW);
            accx = fmaf(wt, vals.x, accx);
            accy = fmaf(wt, vals.y, accy);
        }
        __syncthreads();                                   // done reading buf & s_w
        buf ^= 1;
    }

    const int pidx = b * CHUNKS + chunk;
    part_acc[(size_t)pidx * E + 2 * tid]     = accx;
    part_acc[(size_t)pidx * E + 2 * tid + 1] = accy;
    if (tid == 0) { part_m[pidx] = m_run; part_s[pidx] = s_run; }
}

// ---------------------------------------------------------------------------
// Kernel 2: combine 32 online-softmax partials per batch
// out[b,e] = sum_j exp(m_j - M) * acc_j[e] / (sum_j exp(m_j - M) * s_j)
// ---------------------------------------------------------------------------
__global__ __launch_bounds__(256) void finalize_kernel(const float* __restrict__ part_m,
                                                       const float* __restrict__ part_s,
                                                       const float* __restrict__ part_acc,
                                                       float* __restrict__ out) {
    const int b   = blockIdx.x;
    const int tid = threadIdx.x;
    float M = -INFINITY;
#pragma unroll
    for (int j = 0; j < CHUNKS; ++j) M = fmaxf(M, part_m[b * CHUNKS + j]);
    float S = 0.f, ax = 0.f, ay = 0.f;
    const int e = 2 * tid;
#pragma unroll 4
    for (int j = 0; j < CHUNKS; ++j) {
        const float f = __expf(part_m[b * CHUNKS + j] - M);
        S += f * part_s[b * CHUNKS + j];
        const float* pa = part_acc + (size_t)(b * CHUNKS + j) * E + e;
        ax = fmaf(f, pa[0], ax);
        ay = fmaf(f, pa[1], ay);
    }
    out[b * E + e]     = ax / S;
    out[b * E + e + 1] = ay / S;
}

// ---------------------------------------------------------------------------
extern "C" void kernel_launch(void* const* d_in, const int* in_sizes, int n_in,
                              void* d_out, int out_size, void* d_ws, size_t ws_size,
                              hipStream_t stream) {
    const float* values = (const float*)d_in[0];
    const float* query  = (const float*)d_in[1];
    const float* v      = (const float*)d_in[2];
    const float* W1_w   = (const float*)d_in[3];
    const float* W1_b   = (const float*)d_in[4];
    const float* W2_w   = (const float*)d_in[5];
    const float* W2_b   = (const float*)d_in[6];
    float* out = (float*)d_out;

    // workspace layout (floats): cst[B*D] | part_m[B*32] | part_s[B*32] | part_acc[B*32*E]
    float* ws       = (float*)d_ws;
    float* cst      = ws;
    float* part_m   = cst + B * D;
    float* part_s   = part_m + B * CHUNKS;
    float* part_acc = part_s + B * CHUNKS;

    prep_kernel<<<B, D, 0, stream>>>(query, W1_w, W1_b, W2_b, cst);

    dim3 grid(CHUNKS, B);
    main_kernel<<<grid, 256, 0, stream>>>(values, W2_w, v, cst, part_m, part_s, part_acc);

    finalize_kernel<<<B, 256, 0, stream>>>(part_m, part_s, part_acc, out);
}